// InstanceSegLoss_29437705847445
// MI455X (gfx1250) — compile-verified
//
#include <hip/hip_runtime.h>
#include <math.h>

// ---------------------------------------------------------------------------
// Problem constants (match reference)
// ---------------------------------------------------------------------------
#define B_    16
#define C_    32
#define K_    128
#define G_    32
#define HW_   25600      // 160*160
#define N0_   6400
#define N1_   1600
#define N2_   400
#define NTOT_ 8400
#define EPS_  1e-6f

typedef __attribute__((ext_vector_type(2))) float v2f;
typedef __attribute__((ext_vector_type(8))) float v8f;

// ---------------------------------------------------------------------------
// DPP xor-style reduction across a 16-lane row (lanes 0-15 / 16-31 stay
// separate). Pure VALU: v_mov_b32_dpp + v_add_f32 per step, no DS traffic.
//   step1: quad_perm(1,0,3,2)  == xor 1
//   step2: quad_perm(2,3,0,1)  == xor 2
//   step3: row_half_mirror     == xor 7 (acts as xor 4 once quads uniform)
//   step4: row_mirror          == xor 15 (acts as xor 8 once halves uniform)
// ---------------------------------------------------------------------------
template <int CTRL>
__device__ __forceinline__ float dpp_add(float v) {
    int sh = __builtin_amdgcn_update_dpp(0, __float_as_int(v), CTRL, 0xF, 0xF, true);
    return v + __int_as_float(sh);
}
__device__ __forceinline__ float row_reduce16(float v) {
    v = dpp_add<0xB1>(v);    // quad_perm [1,0,3,2]
    v = dpp_add<0x4E>(v);    // quad_perm [2,3,0,1]
    v = dpp_add<0x141>(v);   // row_half_mirror
    v = dpp_add<0x140>(v);   // row_mirror
    return v;                // all 16 lanes of the row hold the row-sum
}

// ---------------------------------------------------------------------------
// Kernel 0: zero the accumulator region (harness poisons ws once; we must be
// idempotent across graph replays).
// ---------------------------------------------------------------------------
__global__ void zero_accum_kernel(float* __restrict__ accum, int n) {
    int i = blockIdx.x * blockDim.x + threadIdx.x;
    if (i < n) accum[i] = 0.0f;
}

// ---------------------------------------------------------------------------
// Kernel 1: gather positive coefficients (B,K,C) and matched GT index (B,K).
// ---------------------------------------------------------------------------
__global__ void gather_kernel(const float* __restrict__ s0,
                              const float* __restrict__ s1,
                              const float* __restrict__ s2,
                              const int*   __restrict__ pos,
                              const int*   __restrict__ mgt,
                              float* __restrict__ coeff,
                              int*   __restrict__ matched) {
    int t = blockIdx.x * blockDim.x + threadIdx.x;   // B*K*C = 65536 threads
    int c = t & (C_ - 1);
    int k = (t >> 5) & (K_ - 1);
    int b = t >> 12;
    int n = pos[b * K_ + k];
    float v;
    if (n < N0_)            v = s0[(b * C_ + c) * N0_ + n];
    else if (n < N0_ + N1_) v = s1[(b * C_ + c) * N1_ + (n - N0_)];
    else                    v = s2[(b * C_ + c) * N2_ + (n - N0_ - N1_)];
    coeff[(b * K_ + k) * C_ + c] = v;
    if (c == 0) matched[b * K_ + k] = mgt[b * NTOT_ + n];
}

// ---------------------------------------------------------------------------
// Kernel 2: fused WMMA GEMM + sigmoid/softplus BCE + dice partial sums.
//
// Grid: (HW/64, B). Block: 128 threads = 4 wave32; wave w owns a 16-column
// strip. A (128x32 f32, 16KB) staged in LDS per workgroup; each wave keeps
// all 8 B-fragments resident in VGPRs -> prototype_mask read from HBM once.
//
// Per M-tile: issue the 8 gt row loads first (latency hidden under the WMMA
// chain), 8x v_wmma_f32_16x16x4_f32 over C, then a fused epilogue with
// v_rcp-based sigmoid, DPP row reductions and ballot/popc for the gt count.
// ---------------------------------------------------------------------------
__launch_bounds__(128, 1)
__global__ void seg_loss_main(const float* __restrict__ proto,
                              const float* __restrict__ gt,
                              const float* __restrict__ coeff,
                              const int*   __restrict__ matched,
                              float* __restrict__ accum,   // [B*K][3] = inter, sumPred, sumGt
                              float* __restrict__ bce) {   // [1]
    __shared__ float Alds[K_ * C_];
    __shared__ int   Mlds[K_];

    const int b = blockIdx.y;

    for (int i = threadIdx.x; i < K_ * C_; i += 128)
        Alds[i] = coeff[b * K_ * C_ + i];
    if (threadIdx.x < K_)
        Mlds[threadIdx.x] = matched[b * K_ + threadIdx.x];
    __syncthreads();

    const int lane = threadIdx.x & 31;
    const int wv   = threadIdx.x >> 5;
    const int half = lane >> 4;       // 0: lanes 0-15, 1: lanes 16-31
    const int ncol = lane & 15;       // column within tile / M-row for A frag
    const int hw   = blockIdx.x * 64 + wv * 16 + ncol;
    const unsigned halfMask = half ? 0xFFFF0000u : 0x0000FFFFu;

    // Load the full C=32 stack of B-fragments once (8 fragments of 4 K-rows).
    const float* Pb = proto + (b * C_) * HW_;
    v2f bf[8];
#pragma unroll
    for (int j = 0; j < 8; ++j) {
        int cc = 4 * j + 2 * half;
        bf[j].x = Pb[cc * HW_ + hw];
        bf[j].y = Pb[(cc + 1) * HW_ + hw];
    }

    float bceAcc = 0.0f;

    for (int mt = 0; mt < 8; ++mt) {
        // Issue this tile's 8 gt row loads early; WMMA chain hides latency.
        float tvr[8];
#pragma unroll
        for (int v = 0; v < 8; ++v) {
            int inst = mt * 16 + v + 8 * half;
            tvr[v] = gt[(b * G_ + Mlds[inst]) * HW_ + hw];
        }

        // A fragments for this 16-row tile from LDS.
        v2f af[8];
        const float* Ar = &Alds[(mt * 16 + ncol) * C_];
#pragma unroll
        for (int j = 0; j < 8; ++j) {
            int cc = 4 * j + 2 * half;
            af[j].x = Ar[cc];
            af[j].y = Ar[cc + 1];
        }

        // 16x16 logits tile: accumulate over C in chunks of 4.
        v8f acc = {0.f, 0.f, 0.f, 0.f, 0.f, 0.f, 0.f, 0.f};
#pragma unroll
        for (int j = 0; j < 8; ++j)
            acc = __builtin_amdgcn_wmma_f32_16x16x4_f32(
                false, af[j], false, bf[j], (short)0, acc, false, false);

        // Fused epilogue: sigmoid, softplus-BCE, dice partials.
#pragma unroll
        for (int v = 0; v < 8; ++v) {
            float x    = acc[v];
            int   inst = mt * 16 + v + 8 * half;   // instance row (0..127)
            bool  on   = tvr[v] > 0.5f;
            float t    = on ? 1.0f : 0.0f;
            unsigned tm = __builtin_amdgcn_ballot_w32(on);

            float e   = __expf(-fabsf(x));                   // v_exp_f32
            float opc = 1.0f + e;
            float inv = __builtin_amdgcn_rcpf(opc);          // v_rcp_f32
            float p   = (x >= 0.0f ? 1.0f : e) * inv;        // sigmoid
            float sp  = fmaxf(x, 0.0f) + __logf(opc);        // stable softplus
            bceAcc += sp - x * t;

            float ps = row_reduce16(p);
            float is = row_reduce16(p * t);
            if (ncol == 0) {
                float ts = (float)__popc(tm & halfMask);     // exact gt count
                float* a3 = accum + (b * K_ + inst) * 3;
                atomicAdd(a3 + 0, is);
                atomicAdd(a3 + 1, ps);
                atomicAdd(a3 + 2, ts);
            }
        }
    }

    // One BCE atomic per wave: 16-lane DPP reduce + single cross-half shuffle.
    bceAcc = row_reduce16(bceAcc);
    bceAcc += __shfl_xor(bceAcc, 16);
    if (lane == 0) atomicAdd(bce, bceAcc);
}

// ---------------------------------------------------------------------------
// Kernel 3: dice per instance, global mean, combine with BCE -> scalar loss.
// ---------------------------------------------------------------------------
__global__ void finalize_kernel(const float* __restrict__ accum,
                                const float* __restrict__ bce,
                                float* __restrict__ out) {
    __shared__ float red[256];
    float dsum = 0.0f;
    for (int i = threadIdx.x; i < B_ * K_; i += 256) {
        float inter = accum[i * 3 + 0];
        float sp    = accum[i * 3 + 1];
        float st    = accum[i * 3 + 2];
        dsum += (2.0f * inter + EPS_) / (sp + st + EPS_);
    }
    red[threadIdx.x] = dsum;
    __syncthreads();
    for (int s = 128; s > 0; s >>= 1) {
        if (threadIdx.x < s) red[threadIdx.x] += red[threadIdx.x + s];
        __syncthreads();
    }
    if (threadIdx.x == 0) {
        float bceMean = bce[0] / ((float)B_ * (float)K_ * (float)HW_);
        out[0] = bceMean + 1.0f - red[0] / (float)(B_ * K_);
    }
}

// ---------------------------------------------------------------------------
// Launch
// ---------------------------------------------------------------------------
extern "C" void kernel_launch(void* const* d_in, const int* in_sizes, int n_in,
                              void* d_out, int out_size, void* d_ws, size_t ws_size,
                              hipStream_t stream) {
    const float* s0    = (const float*)d_in[0];  // (16,32,6400)
    const float* s1    = (const float*)d_in[1];  // (16,32,1600)
    const float* s2    = (const float*)d_in[2];  // (16,32,400)
    const float* proto = (const float*)d_in[3];  // (16,32,160,160)
    const float* gtm   = (const float*)d_in[4];  // (16,32,160,160)
    const int*   pos   = (const int*)d_in[5];    // (16,128)
    const int*   mgt   = (const int*)d_in[6];    // (16,8400)

    float* ws      = (float*)d_ws;
    float* coeff   = ws;                               // B*K*C   = 65536 f
    int*   matched = (int*)(ws + B_ * K_ * C_);        // B*K     =  2048 i
    float* accum   = ws + B_ * K_ * C_ + B_ * K_;      // B*K*3   =  6144 f
    float* bce     = accum + B_ * K_ * 3;              // 1 f

    const int nAcc = B_ * K_ * 3 + 1;
    zero_accum_kernel<<<(nAcc + 255) / 256, 256, 0, stream>>>(accum, nAcc);

    gather_kernel<<<(B_ * K_ * C_) / 256, 256, 0, stream>>>(
        s0, s1, s2, pos, mgt, coeff, matched);

    seg_loss_main<<<dim3(HW_ / 64, B_), 128, 0, stream>>>(
        proto, gtm, coeff, matched, accum, bce);

    finalize_kernel<<<1, 256, 0, stream>>>(accum, bce, (float*)d_out);
}